// BilateralModule_62921270887097
// MI455X (gfx1250) — compile-verified
//
#include <hip/hip_runtime.h>
#include <stdint.h>

#define IMH 512
#define IMW 512
#define NCH 3
#define NBATCH 16

// ---------------------------------------------------------------------------
// CDNA5 async global->LDS copy (ASYNCcnt path). Builtin arity differs across
// toolchains, so use inline asm (ISA: GLOBAL_LOAD_ASYNC_TO_LDS_B32, §15.18.3
// opcode 96; completion via s_wait_asynccnt).
// The low 32 bits of a generic pointer to LDS are the LDS byte address.
// ---------------------------------------------------------------------------
__device__ __forceinline__ uint32_t lds_lo32(const void* p) {
    return (uint32_t)(uintptr_t)p;
}
__device__ __forceinline__ void async_ld_b32(uint32_t lds_addr, const void* gaddr) {
    asm volatile("global_load_async_to_lds_b32 %0, %1, off"
                 :: "v"(lds_addr), "v"(gaddr) : "memory");
}
__device__ __forceinline__ void wait_async0() {
    asm volatile("s_wait_asynccnt 0x0" ::: "memory");
}

// np.pad 'reflect' (no edge repeat); valid for |overhang| <= 4 << 512
__device__ __forceinline__ int reflect_idx(int q, int n) {
    q = q < 0 ? -q : q;
    q = q > n - 1 ? 2 * n - 2 - q : q;
    return q;
}
__device__ __forceinline__ int clamp_idx(int q, int n) {
    return q < 0 ? 0 : (q > n - 1 ? n - 1 : q);
}

// ---------------------------------------------------------------------------
// Kernel 1: quantize + bilateral filter.  16x16 tile, halo 4 -> 24x24x3 LDS.
// ---------------------------------------------------------------------------
__global__ __launch_bounds__(256) void k_bilateral(const float* __restrict__ x,
                                                   float* __restrict__ filt) {
    __shared__ float tile[NCH][24][24];
    const int bx0 = blockIdx.x * 16, by0 = blockIdx.y * 16, b = blockIdx.z;
    const int tid = threadIdx.y * 16 + threadIdx.x;
    const float* xb = x + (size_t)b * NCH * IMH * IMW;

    // async halo load: 3*24*24 = 1728 words / 256 threads
    for (int k = tid; k < NCH * 24 * 24; k += 256) {
        int c = k / 576, r = k - c * 576, ly = r / 24, lx = r - ly * 24;
        int gy = reflect_idx(by0 - 4 + ly, IMH);
        int gx = reflect_idx(bx0 - 4 + lx, IMW);
        async_ld_b32(lds_lo32(&tile[c][ly][lx]),
                     xb + ((size_t)c * IMH + gy) * IMW + gx);
    }
    wait_async0();
    // quantize in LDS: img = round(clip(x,0,1)*255)  (rint = nearest-even, matches jnp.round)
    for (int k = tid; k < NCH * 24 * 24; k += 256) {
        int c = k / 576, r = k - c * 576, ly = r / 24, lx = r - ly * 24;
        float v = tile[c][ly][lx];
        tile[c][ly][lx] = rintf(fminf(fmaxf(v, 0.0f), 1.0f) * 255.0f);
    }
    __syncthreads();

    const int ty = threadIdx.y, tx = threadIdx.x;
    const float c0 = tile[0][ty + 4][tx + 4];
    const float c1 = tile[1][ty + 4][tx + 4];
    const float c2 = tile[2][ty + 4][tx + 4];
    const float gcc = -0.5f / (75.0f * 75.0f);  // == gs == gc
    float n0 = 0.f, n1 = 0.f, n2 = 0.f, den = 0.f;
#pragma unroll
    for (int dy = -4; dy <= 4; ++dy) {
#pragma unroll
        for (int dx = -4; dx <= 4; ++dx) {
            const int r2 = dy * dy + dx * dx;
            if (r2 > 16) continue;  // disk window (49 taps)
            float s0 = tile[0][ty + 4 + dy][tx + 4 + dx];
            float s1 = tile[1][ty + 4 + dy][tx + 4 + dx];
            float s2 = tile[2][ty + 4 + dy][tx + 4 + dx];
            float cd = fabsf(s0 - c0) + fabsf(s1 - c1) + fabsf(s2 - c2);
            float w = __expf(gcc * (float)r2 + gcc * cd * cd);
            n0 = fmaf(w, s0, n0); n1 = fmaf(w, s1, n1); n2 = fmaf(w, s2, n2);
            den += w;
        }
    }
    const float inv = 1.0f / den;  // den >= 1 (center tap)
    const int y = by0 + ty, xo = bx0 + tx;
    filt[(((size_t)b * NCH + 0) * IMH + y) * IMW + xo] = n0 * inv;
    filt[(((size_t)b * NCH + 1) * IMH + y) * IMW + xo] = n1 * inv;
    filt[(((size_t)b * NCH + 2) * IMH + y) * IMW + xo] = n2 * inv;
}

// ---------------------------------------------------------------------------
// Kernel 2: Sobel (edge-clamped) + channel argmax + directional NMS.
// 16x16 tile, halo 2 -> 20x20x3 filtered LDS; 18x18 mag LDS (zero-padded).
// ---------------------------------------------------------------------------
__device__ __forceinline__ void sobel3(const float (*f)[20][20], int c, int ci, int cj,
                                       float& sgx, float& sgy) {
    float a = f[c][ci - 1][cj - 1], bq = f[c][ci - 1][cj], cc = f[c][ci - 1][cj + 1];
    float d = f[c][ci][cj - 1],     e  = f[c][ci][cj + 1];
    float g = f[c][ci + 1][cj - 1], h  = f[c][ci + 1][cj], i = f[c][ci + 1][cj + 1];
    sgx = (cc + 2.0f * e + i) - (a + 2.0f * d + g);
    sgy = (g + 2.0f * h + i) - (a + 2.0f * bq + cc);
}

__global__ __launch_bounds__(256) void k_nms(const float* __restrict__ filt,
                                             float* __restrict__ nms) {
    __shared__ float f[NCH][20][20];
    __shared__ float mg[18][18];
    const int bx0 = blockIdx.x * 16, by0 = blockIdx.y * 16, b = blockIdx.z;
    const int tid = threadIdx.y * 16 + threadIdx.x;
    const float* fb = filt + (size_t)b * NCH * IMH * IMW;

    for (int k = tid; k < NCH * 20 * 20; k += 256) {
        int c = k / 400, r = k - c * 400, ly = r / 20, lx = r - ly * 20;
        int gy = clamp_idx(by0 - 2 + ly, IMH);  // 'edge' padding
        int gx = clamp_idx(bx0 - 2 + lx, IMW);
        async_ld_b32(lds_lo32(&f[c][ly][lx]),
                     fb + ((size_t)c * IMH + gy) * IMW + gx);
    }
    wait_async0();
    __syncthreads();

    // mag = max_c(|gx_c|+|gy_c|) over tile+1 halo; 0 outside image (jnp.pad(mag,1))
    for (int k = tid; k < 18 * 18; k += 256) {
        int i = k / 18, j = k - 18 * i;            // pixel (by0+i-1, bx0+j-1)
        int gy = by0 + i - 1, gx = bx0 + j - 1;
        float m = 0.0f;
        if (gy >= 0 && gy < IMH && gx >= 0 && gx < IMW) {
#pragma unroll
            for (int c = 0; c < NCH; ++c) {
                float sx, sy;
                sobel3(f, c, i + 1, j + 1, sx, sy);
                float mc = fabsf(sx) + fabsf(sy);
                m = fmaxf(m, mc);
            }
        }
        mg[i][j] = m;
    }
    __syncthreads();

    const int ty = threadIdx.y, tx = threadIdx.x;
    // argmax channel (first-wins on ties, like jnp.argmax)
    float gxv, gyv, best;
    sobel3(f, 0, ty + 2, tx + 2, gxv, gyv);
    best = fabsf(gxv) + fabsf(gyv);
#pragma unroll
    for (int c = 1; c < NCH; ++c) {
        float sx, sy;
        sobel3(f, c, ty + 2, tx + 2, sx, sy);
        float mc = fabsf(sx) + fabsf(sy);
        if (mc > best) { best = mc; gxv = sx; gyv = sy; }
    }
    const float mag = best;
    // normalize direction to [0,180): (gx,gy) ~ (-gx,-gy)
    float ax = gxv, ay = gyv;
    if (ay < 0.0f || (ay == 0.0f && ax < 0.0f)) { ax = -ax; ay = -ay; }
    const float t1 = 0.41421356f;   // tan(22.5 deg)
    const float t2 = 2.41421356f;   // tan(67.5 deg)
    const float aax = fabsf(ax);
    int d1y, d1x, d2y, d2x;
    if (ay < t1 * aax)      { d1y = 0;  d1x = 1;  d2y = 0; d2x = -1; }           // d0
    else if (ay < t2 * aax) {
        if (ax > 0.0f)      { d1y = -1; d1x = 1;  d2y = 1; d2x = -1; }           // d45
        else                { d1y = -1; d1x = -1; d2y = 1; d2x = 1;  }           // d135
    } else                  { d1y = -1; d1x = 0;  d2y = 1; d2x = 0;  }           // d90
    const float na = mg[1 + ty + d1y][1 + tx + d1x];
    const float nb = mg[1 + ty + d2y][1 + tx + d2x];
    const float out = (mag >= na && mag >= nb) ? mag : 0.0f;
    nms[((size_t)b * IMH + by0 + ty) * IMW + bx0 + tx] = out;
}

// ---------------------------------------------------------------------------
// Kernel 3: double threshold -> strong state (float) + weak mask (u8)
// ---------------------------------------------------------------------------
__global__ __launch_bounds__(256) void k_thresh(const float* __restrict__ nms,
                                                float* __restrict__ strong,
                                                unsigned char* __restrict__ weak,
                                                int n) {
    int i = blockIdx.x * 256 + threadIdx.x;
    if (i < n) {
        float v = nms[i];
        strong[i] = v > 150.0f ? 1.0f : 0.0f;
        weak[i]   = v > 50.0f ? (unsigned char)1 : (unsigned char)0;
    }
}

// ---------------------------------------------------------------------------
// Kernel 4: hysteresis sweep. 64x64 tile Gauss-Seidel to local fixpoint
// against a halo snapshot from src; Jacobi ping-pong across launches.
// Monotone 0->1 updates => unique fixpoint, deterministic per launch.
// ---------------------------------------------------------------------------
#define TS 64
__global__ __launch_bounds__(256) void k_hyst(const float* __restrict__ src,
                                              const unsigned char* __restrict__ weak,
                                              float* __restrict__ dst) {
    __shared__ unsigned char st[TS + 2][TS + 2];
    __shared__ unsigned char wk[TS][TS];
    __shared__ int changed;
    const int bx0 = blockIdx.x * TS, by0 = blockIdx.y * TS, b = blockIdx.z;
    const size_t plane = (size_t)b * IMH * IMW;
    const int tid = threadIdx.y * 64 + threadIdx.x;

    for (int k = tid; k < (TS + 2) * (TS + 2); k += 256) {
        int i = k / (TS + 2), j = k - (TS + 2) * i;
        int gy = by0 + i - 1, gx = bx0 + j - 1;
        unsigned char v = 0;
        if (gy >= 0 && gy < IMH && gx >= 0 && gx < IMW)
            v = src[plane + (size_t)gy * IMW + gx] > 0.5f ? 1 : 0;
        st[i][j] = v;
    }
    for (int k = tid; k < TS * TS; k += 256) {
        int i = k / TS, j = k - TS * i;
        wk[i][j] = weak[plane + (size_t)(by0 + i) * IMW + bx0 + j];
    }
    __syncthreads();

    int iter = 0;
    while (true) {
        if (tid == 0) changed = 0;
        __syncthreads();
        for (int r = (int)threadIdx.y; r < TS; r += 4) {
            const int i = r + 1, j = (int)threadIdx.x + 1;
            if (!st[i][j] && wk[r][threadIdx.x]) {
                int nb = st[i - 1][j - 1] | st[i - 1][j] | st[i - 1][j + 1] |
                         st[i][j - 1]                    | st[i][j + 1] |
                         st[i + 1][j - 1] | st[i + 1][j] | st[i + 1][j + 1];
                if (nb) { st[i][j] = 1; changed = 1; }
            }
        }
        __syncthreads();
        if (changed == 0 || ++iter >= 256) break;
        __syncthreads();
    }

    for (int k = tid; k < TS * TS; k += 256) {
        int i = k / TS, j = k - TS * i;
        dst[plane + (size_t)(by0 + i) * IMW + bx0 + j] = st[i + 1][j + 1] ? 1.0f : 0.0f;
    }
}

// ---------------------------------------------------------------------------
// Workspace layout (64 MiB total):
//   [0, 48 MiB)           filtered (dead after k_nms)
//     reuse: [0,16 MiB)   stateB   (ping-pong buffer)
//            [16,20 MiB)  weak mask (u8)
//   [48 MiB, 64 MiB)      nms
// ---------------------------------------------------------------------------
extern "C" void kernel_launch(void* const* d_in, const int* in_sizes, int n_in,
                              void* d_out, int out_size, void* d_ws, size_t ws_size,
                              hipStream_t stream) {
    const float* x = (const float*)d_in[0];
    float* out = (float*)d_out;
    char* ws = (char*)d_ws;

    float* filt          = (float*)(ws);
    float* stateB        = (float*)(ws);                       // aliases filt (after k_nms)
    unsigned char* weak  = (unsigned char*)(ws + (size_t)16 * 1024 * 1024);
    float* nms           = (float*)(ws + (size_t)48 * 1024 * 1024);

    dim3 blk(16, 16);
    dim3 grd(IMW / 16, IMH / 16, NBATCH);
    k_bilateral<<<grd, blk, 0, stream>>>(x, filt);
    k_nms<<<grd, blk, 0, stream>>>(filt, nms);

    const int n = NBATCH * IMH * IMW;
    k_thresh<<<dim3((n + 255) / 256), dim3(256), 0, stream>>>(nms, out, weak, n);

    dim3 gh(IMW / TS, IMH / TS, NBATCH), bh(64, 4);
    const int NITER = 24;  // even: final state lands in d_out
    for (int i = 0; i < NITER; ++i) {
        const float* s = (i & 1) ? (const float*)stateB : (const float*)out;
        float* d       = (i & 1) ? out : stateB;
        k_hyst<<<gh, bh, 0, stream>>>(s, weak, d);
    }
}